// CorrelationLayerCosineSimilarity_13580686590326
// MI455X (gfx1250) — compile-verified
//
#include <hip/hip_runtime.h>
#include <math.h>
#include <stdint.h>

// CDNA5 / gfx1250, wave32
typedef float v2f __attribute__((ext_vector_type(2)));
typedef float v8f __attribute__((ext_vector_type(8)));
typedef int   v4i __attribute__((ext_vector_type(4)));
typedef int   v8i __attribute__((ext_vector_type(8)));

#define BB 4
#define CC 256
#define HH 128
#define WW 256
#define ND 41            // displacements 0..40
#define KC 8             // channels per chunk (double-buffered)
#define NCHUNK (CC / KC) // 32
#define LDA 260          // padded LDS row stride (floats) for x1 tile
#define LDB 308          // padded LDS row stride for x2 tile (256 data + 52 pad)
#define LDBAND 65

// LDS layout (floats), assumed to start at LDS byte offset 0:
//   buf0: A [0,2080)  B [2080,4544)
//   buf1: A [4544,6624) B [6624,9088)
//   band: [0,8320)  (union with buffers; used after compute)
//   normA [9088,9344)  normB [9344,9648)  scratch [9648,10160)
#define BUF_FLOATS 4544
#define BUF_BYTES  18176
#define BOFF_F     2080
#define BOFF_BYTES 8320
#define NORMA_F    9088
#define NORMB_F    9344
#define SCR_F      9648
#define SMEM_FLOATS 10160

// Issue one TDM 2-D tile load: KC rows x 256 floats, row stride H*W floats,
// LDS rows padded via the TDM pad feature (pad_interval=256 dwords).
//   pad_enc: encoded pad amount in dwords-1 (3 -> 4 dwords, 51 -> 52 dwords)
__device__ __forceinline__ void tdm_load_tile(uint32_t lds_byte_off,
                                              const float* gptr,
                                              uint32_t pad_enc)
{
    uint64_t ga = (uint64_t)(uintptr_t)gptr;
    v4i g0;
    g0[0] = 1;                                        // count=1, user mode
    g0[1] = (int)lds_byte_off;                        // lds_addr
    g0[2] = (int)(uint32_t)ga;                        // global_addr[31:0]
    g0[3] = (int)(((uint32_t)(ga >> 32) & 0x01FFFFFFu) | 0x80000000u); // addr[56:32] | type=2
    v8i g1;
    g1[0] = (int)((2u << 16) | (1u << 20) | (7u << 22) | (pad_enc << 25));
    //            data_size=4B  pad_enable  pad_interval=256dw  pad_amount
    g1[1] = (int)(256u << 16);                        // tensor_dim0 = 256 (bits 79:48)
    g1[2] = (int)((uint32_t)KC << 16);                // tensor_dim1 = KC (bits 111:80)
    g1[3] = (int)(256u << 16);                        // tile_dim0 = 256 (bits 127:112)
    g1[4] = (int)KC;                                  // tile_dim1 = KC, tile_dim2 = 0
    g1[5] = (int)(HH * WW);                           // tensor_dim0_stride = 32768
    g1[6] = 0;                                        // dim0_stride hi, dim1_stride lo
    g1[7] = 0;
    asm volatile("tensor_load_to_lds %0, %1" :: "s"(g0), "s"(g1) : "memory");
}

__global__ __launch_bounds__(256)
void corr_cos_wmma_tdm(const float* __restrict__ x1,
                       const float* __restrict__ x2,
                       float* __restrict__ out)
{
    __shared__ float smem[SMEM_FLOATS];

    const int tid  = threadIdx.x;
    const int lane = tid & 31;
    const int wave = tid >> 5;             // 0..7
    const int b    = blockIdx.x >> 7;      // blockIdx = b*H + h
    const int h    = blockIdx.x & (HH - 1);

    const float* base1 = x1 + ((size_t)b * CC * HH + h) * WW;
    const float* base2 = x2 + ((size_t)b * CC * HH + h) * WW;

    // Zero the right-pad columns (cols 256..307) of BOTH x2 buffers once:
    // the TDM pad feature skips them, so dot products vs w' >= W stay 0.
    for (int idx = tid; idx < 2 * KC * (LDB - WW); idx += 256) {
        int bufi = idx / (KC * (LDB - WW));
        int rem  = idx % (KC * (LDB - WW));
        int r    = rem / (LDB - WW);
        int c    = rem % (LDB - WW);
        smem[bufi * BUF_FLOATS + BOFF_F + r * LDB + WW + c] = 0.0f;
    }

    // ---- prologue: TDM-stage chunk 0 into buffer 0
    if (wave == 0) {
        tdm_load_tile(0,          base1, 3);   // A: pad 4 dwords  -> stride 260
        tdm_load_tile(BOFF_BYTES, base2, 51);  // B: pad 52 dwords -> stride 308
    }

    const int w0a   = wave * 32;
    const int w0b   = w0a + 16;
    const int m     = lane & 15;
    const int khalf = (lane >> 4) << 1;    // lanes 16-31 hold K+2,K+3 (A/B frags)
    const int chalf = (lane >> 4) << 3;    // lanes 16-31 hold rows M+8 (C/D)

    // norm-pass ownership: 4-column group of one input / one row-half
    const int g    = tid & 63;             // columns 4g..4g+3
    const int half = (tid >> 6) & 1;       // rows half*4 .. half*4+3
    const int isB  = tid >> 7;             // 0: x1 norms, 1: x2 norms

    v8f accA[4] = {};
    v8f accB[4] = {};
    float n0 = 0.f, n1 = 0.f, n2 = 0.f, n3 = 0.f;

    for (int kc = 0; kc < NCHUNK; ++kc) {
        __builtin_amdgcn_s_wait_tensorcnt(0);   // chunk kc landed (wave0; no-op elsewhere)
        __syncthreads();                        // visible to all waves

        // prefetch chunk kc+1 into the other buffer (its readers passed the
        // previous end-of-iteration barrier)
        if (wave == 0 && kc + 1 < NCHUNK) {
            const size_t coff = (size_t)(kc + 1) * KC * (HH * WW);
            const uint32_t nb = (uint32_t)((kc + 1) & 1) * BUF_BYTES;
            tdm_load_tile(nb,              base1 + coff, 3);
            tdm_load_tile(nb + BOFF_BYTES, base2 + coff, 51);
        }

        const float* cA = smem + (kc & 1) * BUF_FLOATS;
        const float* cB = cA + BOFF_F;

        // ---- fused norm accumulation (vectorized, LDS-resident data)
        {
            const int ld = isB ? LDB : LDA;
            const float* src = (isB ? cB : cA) + (half * 4) * ld + 4 * g;
            #pragma unroll
            for (int r = 0; r < 4; ++r) {
                float4 v = *(const float4*)(src + r * ld);
                n0 = fmaf(v.x, v.x, n0); n1 = fmaf(v.y, v.y, n1);
                n2 = fmaf(v.z, v.z, n2); n3 = fmaf(v.w, v.w, n3);
            }
        }

        // ---- WMMA band GEMM: 2 k-steps x 8 wmma (2 M-tiles x 4 N-tiles)
        #pragma unroll
        for (int ks = 0; ks < KC / 4; ++ks) {
            const float* rA0 = cA + (ks * 4 + khalf) * LDA;
            const float* rA1 = rA0 + LDA;
            const float* rB0 = cB + (ks * 4 + khalf) * LDB;
            const float* rB1 = rB0 + LDB;

            v2f aA; aA.x = rA0[w0a + m]; aA.y = rA1[w0a + m];
            v2f aB; aB.x = rA0[w0b + m]; aB.y = rA1[w0b + m];

            v2f bf[5];
            #pragma unroll
            for (int j = 0; j < 5; ++j) {
                int n = w0a + 16 * j + m;
                bf[j].x = rB0[n]; bf[j].y = rB1[n];
            }

            #pragma unroll
            for (int t = 0; t < 4; ++t) {
                accA[t] = __builtin_amdgcn_wmma_f32_16x16x4_f32(
                    false, aA, false, bf[t],     (short)0, accA[t], false, false);
                accB[t] = __builtin_amdgcn_wmma_f32_16x16x4_f32(
                    false, aB, false, bf[t + 1], (short)0, accB[t], false, false);
            }
        }

        asm volatile("s_wait_dscnt 0" ::: "memory");  // our LDS reads retired
        __syncthreads();                              // buffer may be overwritten
    }

    // ---- norm finalization: combine the two row-half partials, sqrt
    float* scratch = smem + SCR_F;
    float* normA   = smem + NORMA_F;
    float* normB   = smem + NORMB_F;
    if (half == 1) {
        *(float4*)(scratch + (isB * 64 + g) * 4) = make_float4(n0, n1, n2, n3);
    }
    if (tid < 48) normB[WW + tid] = 0.0f;   // norms of zero-padded columns
    __syncthreads();
    if (half == 0) {
        float4 p = *(const float4*)(scratch + (isB * 64 + g) * 4);
        float4 r;
        r.x = sqrtf(n0 + p.x); r.y = sqrtf(n1 + p.y);
        r.z = sqrtf(n2 + p.z); r.w = sqrtf(n3 + p.w);
        *(float4*)((isB ? normB : normA) + 4 * g) = r;
    }
    __syncthreads();                        // compute done; band may reuse buffers

    // ---- band transpose through LDS + coalesced emit
    float* bw = smem + wave * (16 * LDBAND);

    #pragma unroll
    for (int q = 0; q < 2; ++q) {
        const int w0 = q ? w0b : w0a;
        v8f* acc = q ? accB : accA;

        #pragma unroll
        for (int t = 0; t < 4; ++t) {
            #pragma unroll
            for (int r = 0; r < 8; ++r) {
                bw[(r + chalf) * LDBAND + t * 16 + m] = acc[t][r];
            }
        }
        __syncthreads();

        for (int p = 0; p < 21; ++p) {
            int d = p * 2 + (lane >> 4);
            if (d <= 40) {
                float v   = bw[m * LDBAND + m + d];
                float den = fmaxf(normA[w0 + m] * normB[w0 + m + d], 1e-6f);
                out[(((size_t)b * ND + d) * HH + h) * WW + w0 + m] = v / den;
            }
        }
        __syncthreads();
    }
}

extern "C" void kernel_launch(void* const* d_in, const int* in_sizes, int n_in,
                              void* d_out, int out_size, void* d_ws, size_t ws_size,
                              hipStream_t stream) {
    (void)in_sizes; (void)n_in; (void)out_size; (void)d_ws; (void)ws_size;
    const float* x1 = (const float*)d_in[0];
    const float* x2 = (const float*)d_in[1];
    float* out = (float*)d_out;
    corr_cos_wmma_tdm<<<dim3(BB * HH), dim3(256), 0, stream>>>(x1, x2, out);
}